// MultiHeadAttention_24893630447703
// MI455X (gfx1250) — compile-verified
//
#include <hip/hip_runtime.h>
#include <math.h>

// Problem constants (B,S,D,H fixed by the reference)
#define B_  2
#define S_  2048
#define D_  1024
#define H_  16
#define DH_ 64
#define SCALE_ 0.03125f   // 1/sqrt(1024)

typedef float v2f __attribute__((ext_vector_type(2)));
typedef float v8f __attribute__((ext_vector_type(8)));
typedef int   v4i_ __attribute__((__vector_size__(16)));   // matches builtin param

// ---------------------------------------------------------------------------
// CDNA5 async global->LDS copy (ASYNCcnt path), with synchronous fallback so
// the file compiles on toolchains lacking the builtin.
// ---------------------------------------------------------------------------
#if defined(__has_builtin)
#if __has_builtin(__builtin_amdgcn_global_load_async_to_lds_b128) && \
    __has_builtin(__builtin_amdgcn_s_wait_asynccnt)
#define USE_ASYNC 1
#endif
#endif
#ifndef USE_ASYNC
#define USE_ASYNC 0
#endif

__device__ __forceinline__ void cp16(float* ldst, const float* gsrc) {
#if USE_ASYNC
  __builtin_amdgcn_global_load_async_to_lds_b128(
      (__attribute__((address_space(1))) v4i_*)gsrc,
      (__attribute__((address_space(3))) v4i_*)ldst, 0, 0);
#else
  *(float4*)ldst = *(const float4*)gsrc;
#endif
}

#if USE_ASYNC
#define ASYNC_WAIT(n) __builtin_amdgcn_s_wait_asynccnt(n)
#else
#define ASYNC_WAIT(n) do {} while (0)
#endif

// D = A(16x4) * B(4x16) + C, full fp32 (V_WMMA_F32_16X16X4_F32)
__device__ __forceinline__ v8f wmma_f32(v2f a, v2f b, v8f c) {
  return __builtin_amdgcn_wmma_f32_16x16x4_f32(false, a, false, b, (short)0, c,
                                               false, false);
}

// ---------------------------------------------------------------------------
// Tiled GEMM + bias: C[M,N] = A[M,K] @ W[K,N] + bias[N]   (fp32, WMMA)
// 256 threads (8 waves), 128x64 WG tile, wave grid 4(M) x 2(N): each wave owns
// a 32x32 register block (2x2 WMMA tiles -> 2x A/B fragment reuse).
// K staged 32-wide, double-buffered via async global->LDS copies.
// ---------------------------------------------------------------------------
__global__ __launch_bounds__(256)
void gemm_bias_kernel(const float* __restrict__ A, const float* __restrict__ W,
                      const float* __restrict__ bias, float* __restrict__ C,
                      int M, int N, int K)
{
  __shared__ float As[2][128][36];   // [m][k], rows 144B-aligned, banks 4*(9c%16)
  __shared__ float Ws[2][32][68];    // [k][n], rows 272B-aligned

  const int tid  = threadIdx.x;
  const int lane = tid & 31;
  const int wave = tid >> 5;
  const int col  = lane & 15;          // M(A)/N(B) index inside fragment
  const int hi   = (lane >> 4) << 1;   // K offset 0/2 per half-wave
  const int rb   = (lane >> 4) << 3;   // C/D row offset 0/8
  const int wm   = (wave & 3) << 5;    // 0,32,64,96
  const int wn   = (wave >> 2) << 5;   // 0,32
  const int m0 = blockIdx.y * 128;
  const int n0 = blockIdx.x * 64;

  // ---- prologue: stage K-tile 0 into buffer 0 ----
#pragma unroll
  for (int i = 0; i < 4; ++i) {                      // A: 128x32 = 1024 chunks
    int c = tid + i * 256;
    int row = c >> 3, kc = (c & 7) << 2;
    cp16(&As[0][row][kc], &A[(size_t)(m0 + row) * K + kc]);
  }
#pragma unroll
  for (int i = 0; i < 2; ++i) {                      // W: 32x64 = 512 chunks
    int c = tid + i * 256;
    int row = c >> 4, nc = (c & 15) << 2;
    cp16(&Ws[0][row][nc], &W[(size_t)row * N + n0 + nc]);
  }

  v8f acc[4] = {};
  int buf = 0;
  for (int kk = 0; kk < K; kk += 32) {
    if (kk + 32 < K) {                               // prefetch next K-tile
#pragma unroll
      for (int i = 0; i < 4; ++i) {
        int c = tid + i * 256;
        int row = c >> 3, kc = (c & 7) << 2;
        cp16(&As[buf ^ 1][row][kc], &A[(size_t)(m0 + row) * K + kk + 32 + kc]);
      }
#pragma unroll
      for (int i = 0; i < 2; ++i) {
        int c = tid + i * 256;
        int row = c >> 4, nc = (c & 15) << 2;
        cp16(&Ws[buf ^ 1][row][nc], &W[(size_t)(kk + 32 + row) * N + n0 + nc]);
      }
      ASYNC_WAIT(6);      // only the 6 prefetch chunks may remain in flight
    } else {
      ASYNC_WAIT(0);
    }
    __syncthreads();

#pragma unroll
    for (int k0 = 0; k0 < 32; k0 += 4) {
      v2f a0, a1, b0, b1;
      a0.x = As[buf][wm + col][k0 + hi];
      a0.y = As[buf][wm + col][k0 + hi + 1];
      a1.x = As[buf][wm + 16 + col][k0 + hi];
      a1.y = As[buf][wm + 16 + col][k0 + hi + 1];
      b0.x = Ws[buf][k0 + hi + 0][wn + col];
      b0.y = Ws[buf][k0 + hi + 1][wn + col];
      b1.x = Ws[buf][k0 + hi + 0][wn + 16 + col];
      b1.y = Ws[buf][k0 + hi + 1][wn + 16 + col];
      acc[0] = wmma_f32(a0, b0, acc[0]);
      acc[1] = wmma_f32(a0, b1, acc[1]);
      acc[2] = wmma_f32(a1, b0, acc[2]);
      acc[3] = wmma_f32(a1, b1, acc[3]);
    }
    __syncthreads();      // buffer safe to overwrite next iteration
    buf ^= 1;
  }

#pragma unroll
  for (int i = 0; i < 4; ++i) {
    const int tm = wm + ((i >> 1) << 4);
    const int tn = wn + ((i & 1) << 4);
    const float bv = bias[n0 + tn + col];
#pragma unroll
    for (int r = 0; r < 8; ++r) {
      C[(size_t)(m0 + tm + rb + r) * N + (n0 + tn + col)] = acc[i][r] + bv;
    }
  }
}

// ---------------------------------------------------------------------------
// Head repack. Reference does transpose(0,2,1).reshape(B,H,S,DH) for Q,K:
//   qh[b,h,s,e] = q[b, (64s+e)&2047, 64h + ((64s+e)>>11)]
// and a plain reshape for V.
// ---------------------------------------------------------------------------
__global__ __launch_bounds__(256)
void repack_qk_kernel(const float* __restrict__ q, const float* __restrict__ k,
                      float* __restrict__ qh, float* __restrict__ kh)
{
  int idx = blockIdx.x * 256 + threadIdx.x;      // over B*H*S*DH = 1<<22
  int b  = idx >> 21;
  int h  = (idx >> 17) & 15;
  int t  = idx & 131071;                         // 64*s + e within head
  int s2 = t & 2047;
  int d  = (h << 6) + (t >> 11);
  size_t src = (size_t)b * (S_ * D_) + (size_t)s2 * D_ + d;
  qh[idx] = q[src];
  kh[idx] = k[src];
}

__global__ __launch_bounds__(256)
void repack_v_kernel(const float* __restrict__ v, float* __restrict__ vh)
{
  int idx = blockIdx.x * 256 + threadIdx.x;
  int b = idx >> 21;
  int f = idx & ((1 << 21) - 1);                 // h*131072 + 64*s + e
  vh[idx] = v[(size_t)b * (S_ * D_) + (size_t)(f >> 10) * D_ + (f & 1023)];
}

// vsuf[b,h,s,e] = sum_{j>s} vh[b,h,j,e]  (rank-1 correction for the
// reference's post-softmax -1e9 masking). One thread per (b,h,e).
__global__ __launch_bounds__(256)
void vsuffix_kernel(const float* __restrict__ vh, float* __restrict__ vsuf)
{
  int idx = blockIdx.x * 256 + threadIdx.x;      // B*H*DH = 2048
  int e  = idx & 63;
  int bh = idx >> 6;
  const float* src = vh   + (size_t)bh * (S_ * DH_) + e;
  float*       dst = vsuf + (size_t)bh * (S_ * DH_) + e;
  float acc = 0.f;
  for (int s = S_ - 1; s >= 0; --s) {
    dst[(size_t)s * DH_] = acc;
    acc += src[(size_t)s * DH_];
  }
}

// ---------------------------------------------------------------------------
// Attention: one WG = 128 q-rows of one (b,h). 8 waves x 16-row bands.
// Pass 1: full-row softmax stats over all 2048 keys (reference softmaxes
//         BEFORE masking), K tiles double-buffered via async copies.
// Pass 2: causal exp -> per-wave LDS P tile -> P@V WMMA, K/V double-buffered.
// Epilogue: o/l - 1e9 * vsuf, scattered into the [B,S,D] reshape layout.
// ---------------------------------------------------------------------------
__global__ __launch_bounds__(256)
void attn_kernel(const float* __restrict__ qh, const float* __restrict__ kh,
                 const float* __restrict__ vh, const float* __restrict__ vsuf,
                 float* __restrict__ att)
{
  __shared__ float Qs[128][68];
  __shared__ float Ks[2][64][68];
  __shared__ float Vs[2][64][68];
  __shared__ float Ps[8][16][68];   // per-wave P tile (no cross-wave sharing)

  const int tid  = threadIdx.x;
  const int lane = tid & 31;
  const int wave = tid >> 5;
  const int col  = lane & 15;
  const int hi   = (lane >> 4) << 1;
  const int rb   = (lane >> 4) << 3;
  const int wm   = wave << 4;             // wave's 16-row band in the 128 tile

  const int q0 = blockIdx.x << 7;
  const int h  = blockIdx.y;
  const int b  = blockIdx.z;
  const size_t base = ((size_t)b * H_ + h) * (size_t)(S_ * DH_);

  // ---- prologue: Q tile (128x64) + first K tile ----
#pragma unroll
  for (int i = 0; i < 8; ++i) {
    int c = tid + i * 256;
    int row = c >> 4, cc = (c & 15) << 2;
    cp16(&Qs[row][cc], &qh[base + (size_t)(q0 + row) * DH_ + cc]);
  }
#pragma unroll
  for (int i = 0; i < 4; ++i) {
    int c = tid + i * 256;
    int row = c >> 4, cc = (c & 15) << 2;
    cp16(&Ks[0][row][cc], &kh[base + (size_t)row * DH_ + cc]);
  }

  float m8[8], l8[8];
#pragma unroll
  for (int r = 0; r < 8; ++r) { m8[r] = -3.0e38f; l8[r] = 0.f; }

  // ---- pass 1: streaming full-row max / sumexp ----
  int kb = 0;
  for (int jt = 0; jt < S_; jt += 64) {
    if (jt + 64 < S_) {
#pragma unroll
      for (int i = 0; i < 4; ++i) {
        int c = tid + i * 256;
        int row = c >> 4, cc = (c & 15) << 2;
        cp16(&Ks[kb ^ 1][row][cc], &kh[base + (size_t)(jt + 64 + row) * DH_ + cc]);
      }
      ASYNC_WAIT(4);
    } else {
      ASYNC_WAIT(0);
    }
    __syncthreads();

    v8f sc[4] = {};
#pragma unroll
    for (int k0 = 0; k0 < DH_; k0 += 4) {
      v2f a;                                        // A frag reused 4x
      a.x = Qs[wm + col][k0 + hi];
      a.y = Qs[wm + col][k0 + hi + 1];
#pragma unroll
      for (int jsub = 0; jsub < 4; ++jsub) {
        v2f bb;                                     // B = K^T: B[e][j]=Ks[j][e]
        bb.x = Ks[kb][(jsub << 4) + col][k0 + hi];
        bb.y = Ks[kb][(jsub << 4) + col][k0 + hi + 1];
        sc[jsub] = wmma_f32(a, bb, sc[jsub]);
      }
    }

#pragma unroll
    for (int r = 0; r < 8; ++r) {
      float tm_ = fmaxf(fmaxf(sc[0][r], sc[1][r]), fmaxf(sc[2][r], sc[3][r])) * SCALE_;
      tm_ = fmaxf(tm_, __shfl_xor(tm_, 1, 32));     // reduce within 16-lane half
      tm_ = fmaxf(tm_, __shfl_xor(tm_, 2, 32));
      tm_ = fmaxf(tm_, __shfl_xor(tm_, 4, 32));
      tm_ = fmaxf(tm_, __shfl_xor(tm_, 8, 32));
      float mnew = fmaxf(m8[r], tm_);
      float ss = __expf(sc[0][r] * SCALE_ - mnew) + __expf(sc[1][r] * SCALE_ - mnew)
               + __expf(sc[2][r] * SCALE_ - mnew) + __expf(sc[3][r] * SCALE_ - mnew);
      ss += __shfl_xor(ss, 1, 32);
      ss += __shfl_xor(ss, 2, 32);
      ss += __shfl_xor(ss, 4, 32);
      ss += __shfl_xor(ss, 8, 32);
      l8[r] = l8[r] * __expf(m8[r] - mnew) + ss;
      m8[r] = mnew;
    }
    __syncthreads();
    kb ^= 1;
  }

  // ---- pass 2: causal P@V, K/V double-buffered ----
  v8f oacc[4] = {};
  const int jt_end = q0 + 128;
#pragma unroll
  for (int i = 0; i < 4; ++i) {
    int c = tid + i * 256;
    int row = c >> 4, cc = (c & 15) << 2;
    cp16(&Ks[0][row][cc], &kh[base + (size_t)row * DH_ + cc]);
    cp16(&Vs[0][row][cc], &vh[base + (size_t)row * DH_ + cc]);
  }
  kb = 0;
  for (int jt = 0; jt < jt_end; jt += 64) {
    if (jt + 64 < jt_end) {
#pragma unroll
      for (int i = 0; i < 4; ++i) {
        int c = tid + i * 256;
        int row = c >> 4, cc = (c & 15) << 2;
        cp16(&Ks[kb ^ 1][row][cc], &kh[base + (size_t)(jt + 64 + row) * DH_ + cc]);
        cp16(&Vs[kb ^ 1][row][cc], &vh[base + (size_t)(jt + 64 + row) * DH_ + cc]);
      }
      ASYNC_WAIT(8);
    } else {
      ASYNC_WAIT(0);
    }
    __syncthreads();

    v8f sc[4] = {};
#pragma unroll
    for (int k0 = 0; k0 < DH_; k0 += 4) {
      v2f a;
      a.x = Qs[wm + col][k0 + hi];
      a.y = Qs[wm + col][k0 + hi + 1];
#pragma unroll
      for (int jsub = 0; jsub < 4; ++jsub) {
        v2f bb;
        bb.x = Ks[kb][(jsub << 4) + col][k0 + hi];
        bb.y = Ks[kb][(jsub << 4) + col][k0 + hi + 1];
        sc[jsub] = wmma_f32(a, bb, sc[jsub]);
      }
    }

    // p = exp(score*SCALE - m), causal-zeroed; stash 16x64 P in per-wave LDS
#pragma unroll
    for (int jsub = 0; jsub < 4; ++jsub) {
      const int gcol = jt + (jsub << 4) + col;
#pragma unroll
      for (int r = 0; r < 8; ++r) {
        const int grow = q0 + wm + rb + r;
        float p = __expf(sc[jsub][r] * SCALE_ - m8[r]);
        if (gcol > grow) p = 0.f;                  // masked after softmax
        Ps[wave][rb + r][(jsub << 4) + col] = p;
      }
    }

    // oacc += P(16x64) @ V(64x64)
#pragma unroll
    for (int k0 = 0; k0 < 64; k0 += 4) {
      v2f a;                                        // A frag reused 4x
      a.x = Ps[wave][col][k0 + hi];
      a.y = Ps[wave][col][k0 + hi + 1];
#pragma unroll
      for (int esub = 0; esub < 4; ++esub) {
        v2f bb;
        bb.x = Vs[kb][k0 + hi + 0][(esub << 4) + col];
        bb.y = Vs[kb][k0 + hi + 1][(esub << 4) + col];
        oacc[esub] = wmma_f32(a, bb, oacc[esub]);
      }
    }
    __syncthreads();
    kb ^= 1;
  }

  // ---- epilogue: normalize, add -1e9 * suffix(V), scatter to [B,S,D] ----
#pragma unroll
  for (int esub = 0; esub < 4; ++esub) {
#pragma unroll
    for (int r = 0; r < 8; ++r) {
      const int grow = q0 + wm + rb + r;          // s
      const int e    = (esub << 4) + col;
      float val = oacc[esub][r] / l8[r]
                - 1.0e9f * vsuf[base + (size_t)grow * DH_ + e];
      const int t2 = grow * DH_ + e;              // reshape: f = h*131072 + t2
      const int s_out = (h << 7) + (t2 >> 10);
      const int d_out = t2 & 1023;
      att[(size_t)b * (S_ * D_) + (size_t)s_out * D_ + d_out] = val;
    }
  }
}

// ---------------------------------------------------------------------------
extern "C" void kernel_launch(void* const* d_in, const int* in_sizes, int n_in,
                              void* d_out, int out_size, void* d_ws, size_t ws_size,
                              hipStream_t stream)
{
  const float* Q  = (const float*)d_in[0];
  const float* K  = (const float*)d_in[1];
  const float* V  = (const float*)d_in[2];
  // d_in[3] = mask: causal triu(k=1), handled analytically (never read)
  const float* Wq = (const float*)d_in[4];
  const float* bq = (const float*)d_in[5];
  const float* Wk = (const float*)d_in[6];
  const float* bk = (const float*)d_in[7];
  const float* Wv = (const float*)d_in[8];
  const float* bv = (const float*)d_in[9];
  const float* Wo = (const float*)d_in[10];
  const float* bo = (const float*)d_in[11];
  float* out = (float*)d_out;

  const size_t CH = (size_t)B_ * S_ * D_;   // 4 Mi floats per tensor
  float* ws   = (float*)d_ws;
  float* q    = ws + 0 * CH;
  float* k    = ws + 1 * CH;
  float* v    = ws + 2 * CH;
  float* qh   = ws + 3 * CH;
  float* kh   = ws + 4 * CH;
  float* vh   = ws + 5 * CH;
  float* vsuf = k;    // k is dead after repack
  float* attb = q;    // q is dead after repack

  dim3 blk(256);
  dim3 ggrid(D_ / 64, (B_ * S_) / 128);     // (16, 32)

  gemm_bias_kernel<<<ggrid, blk, 0, stream>>>(Q, Wq, bq, q, B_ * S_, D_, D_);
  gemm_bias_kernel<<<ggrid, blk, 0, stream>>>(K, Wk, bk, k, B_ * S_, D_, D_);
  gemm_bias_kernel<<<ggrid, blk, 0, stream>>>(V, Wv, bv, v, B_ * S_, D_, D_);

  repack_qk_kernel<<<dim3((B_ * H_ * S_ * DH_) / 256), blk, 0, stream>>>(q, k, qh, kh);
  repack_v_kernel <<<dim3((B_ * H_ * S_ * DH_) / 256), blk, 0, stream>>>(v, vh);
  vsuffix_kernel  <<<dim3((B_ * H_ * DH_) / 256), blk, 0, stream>>>(vh, vsuf);

  attn_kernel<<<dim3(S_ / 128, H_, B_), blk, 0, stream>>>(qh, kh, vh, vsuf, attb);

  gemm_bias_kernel<<<ggrid, blk, 0, stream>>>(attb, Wo, bo, out, B_ * S_, D_, D_);

  (void)in_sizes; (void)n_in; (void)out_size; (void)ws_size;
}